// Yolo_loss_30210799960245
// MI455X (gfx1250) — compile-verified
//
#include <hip/hip_runtime.h>
#include <hip/hip_bf16.h>
#include <math.h>

typedef float v2f __attribute__((ext_vector_type(2)));
typedef float v8f __attribute__((ext_vector_type(8)));

#define NB       32
#define NA_      3
#define NH       52
#define NW       52
#define NC_      80
#define G_       20
#define CH       85            // 5 + NC
#define PLANE    (NH*NW)       // 2704
#define BSTRIDE  (NA_*CH*PLANE) // 689520 floats per batch
#define CELLS_PER_B (NA_*PLANE) // 8112
#define NBLK2    32            // ceil(8112/256)
#define TOT2     (NBLK2*NB)    // 1024 block partials
#define NBLK3    3

__device__ __forceinline__ float sigmoidf_(float x) { return 1.0f / (1.0f + __expf(-x)); }
__device__ __forceinline__ float softplusf_(float x) { return fmaxf(x, 0.0f) + log1pf(__expf(-fabsf(x))); }
__device__ __forceinline__ float anchW(int a) { return a == 0 ? 1.25f : (a == 1 ? 2.0f : 4.125f); }
__device__ __forceinline__ float anchH(int a) { return a == 0 ? 1.625f : (a == 1 ? 3.75f : 2.875f); }

// Exact f32 full-wave (32-lane) sum via the matrix pipe:
// A (16x4 f32): VGPR0 = v (K=0 for lanes 0-15, K=2 for lanes 16-31), VGPR1 = 0.
// B = all ones  =>  D[m][n] = v[m] + v[m+16] for every n.
// Lane L sums its 8 D VGPRs (M = 0..7 or 8..15 at column N=L%16), then one
// shfl_xor(16) completes the 16-row sum. Result broadcast to all lanes.
__device__ __forceinline__ float wave_sum(float v) {
#if __has_builtin(__builtin_amdgcn_wmma_f32_16x16x4_f32)
  v2f a; a.x = v;    a.y = 0.0f;
  v2f b; b.x = 1.0f; b.y = 1.0f;
  v8f c = {};
  c = __builtin_amdgcn_wmma_f32_16x16x4_f32(false, a, false, b, (short)0, c, false, false);
  float s = c[0] + c[1] + c[2] + c[3] + c[4] + c[5] + c[6] + c[7];
  s += __shfl_xor(s, 16, 32);
  return s;
#else
  for (int off = 16; off > 0; off >>= 1) v += __shfl_xor(v, off, 32);
  return v;
#endif
}

// ---------------- Kernel 1: GT preprocessing (640 entries) ----------------
__global__ void k_prep(const float* __restrict__ bb, const int* __restrict__ lab,
                       const int* __restrict__ aidx,
                       float* __restrict__ gtbox, float* __restrict__ gtt,
                       int* __restrict__ gmeta, float* __restrict__ glast) {
  int e = blockIdx.x * blockDim.x + threadIdx.x;
  if (e >= NB * G_) return;
  int b = e / G_, g = e - b * G_;
  float bx = bb[e*4+0] * 0.125f, by = bb[e*4+1] * 0.125f;  // /SCALE, SCALE=8
  float bw = bb[e*4+2] * 0.125f, bh = bb[e*4+3] * 0.125f;
  int a = aidx[e];
  int gi = min(max((int)bx, 0), NW - 1);
  int gj = min(max((int)by, 0), NH - 1);
  gtbox[e*5+0] = bx - bw*0.5f;
  gtbox[e*5+1] = by - bh*0.5f;
  gtbox[e*5+2] = bx + bw*0.5f;
  gtbox[e*5+3] = by + bh*0.5f;
  gtbox[e*5+4] = bw * bh;
  gtt[e*4+0] = bx - (float)gi;
  gtt[e*4+1] = by - (float)gj;
  gtt[e*4+2] = logf(bw / anchW(a));
  gtt[e*4+3] = logf(bh / anchH(a));
  gmeta[e*4+0] = gi; gmeta[e*4+1] = gj; gmeta[e*4+2] = a; gmeta[e*4+3] = lab[e];
  // scatter "last index wins": entry is live only if no later g maps to same cell
  float last = 1.0f;
  for (int g2 = g + 1; g2 < G_; ++g2) {
    int e2 = b * G_ + g2;
    float bx2 = bb[e2*4+0] * 0.125f, by2 = bb[e2*4+1] * 0.125f;
    int a2 = aidx[e2];
    int gi2 = min(max((int)bx2, 0), NW - 1), gj2 = min(max((int)by2, 0), NH - 1);
    if (a2 == a && gi2 == gi && gj2 == gj) last = 0.0f;
  }
  glast[e] = last;
}

// ---------- Kernel 2: per-cell streaming pass (only the 5 live channels) ----------
__global__ void k_main(const float* __restrict__ outp, const int* __restrict__ warmp,
                       const float* __restrict__ gtbox, float* __restrict__ p2) {
  __shared__ float sgt[G_*5];
  __shared__ float swv[8][5];
  int b = blockIdx.y;
  int tid = threadIdx.x;
  if (tid < G_*5) sgt[tid] = gtbox[b*G_*5 + tid];
  __syncthreads();
  int warm = warmp[0];
  int t = blockIdx.x * 256 + tid;
  float valid = (t < CELLS_PER_B) ? 1.0f : 0.0f;
  int tc = t < CELLS_PER_B ? t : CELLS_PER_B - 1;      // clamp: keep EXEC full
  int a = tc / PLANE;
  int r = tc - a * PLANE;
  int j = r / NW;
  int i = r - j * NW;
  const float* base = outp + (size_t)b*BSTRIDE + (size_t)a*CH*PLANE + r;
  float x = base[0], y = base[PLANE], w = base[2*PLANE], h = base[3*PLANE], conf = base[4*PLANE];
  float sx = sigmoidf_(x), sy = sigmoidf_(y);
  float pw = __expf(w) * anchW(a), ph = __expf(h) * anchH(a);
  float px = sx + (float)i, py = sy + (float)j;
  float p0 = px - pw*0.5f, p1 = py - ph*0.5f, p2c = px + pw*0.5f, p3c = py + ph*0.5f;
  float ap = pw * ph;
  float miou = 0.0f;
#pragma unroll
  for (int g = 0; g < G_; ++g) {
    float g0 = sgt[g*5+0], g1 = sgt[g*5+1], g2 = sgt[g*5+2], g3 = sgt[g*5+3], ag = sgt[g*5+4];
    float iw = fmaxf(fminf(p2c, g2) - fmaxf(p0, g0), 0.0f);
    float ih = fmaxf(fminf(p3c, g3) - fmaxf(p1, g1), 0.0f);
    float inter = iw * ih;
    miou = fmaxf(miou, inter / (ap + ag - inter + 1e-10f));
  }
  float cw = (miou > 0.5f) ? 0.0f : 0.5f;               // NOOBJ_SCALE / ignore
  float c_conf = valid * cw * softplusf_(conf);         // bce(conf, 0) = softplus
  float c_xc = 0.f, c_yc = 0.f, c_w = 0.f, c_h = 0.f;
  if (warm) {                                           // wave-uniform branch
    c_xc = valid * (sx - 0.5f) * (sx - 0.5f);
    c_yc = valid * (sy - 0.5f) * (sy - 0.5f);
    c_w  = valid * w * w;
    c_h  = valid * h * h;
  }
  float s0 = wave_sum(c_xc), s1 = wave_sum(c_yc), s2 = wave_sum(c_w),
        s3 = wave_sum(c_h),  s4 = wave_sum(c_conf);
  int wv = tid >> 5;
  if ((tid & 31) == 0) { swv[wv][0]=s0; swv[wv][1]=s1; swv[wv][2]=s2; swv[wv][3]=s3; swv[wv][4]=s4; }
  __syncthreads();
  if (tid == 0) {
    int blk = b * NBLK2 + blockIdx.x;
    for (int cmp = 0; cmp < 5; ++cmp) {
      float s = 0.f;
      for (int k = 0; k < 8; ++k) s += swv[k][cmp];
      p2[cmp*TOT2 + blk] = s;
    }
  }
}

// ---------- Kernel 3: 640 matched-cell corrections + cls BCE + nCorrect ----------
__global__ void k_match(const float* __restrict__ outp, const int* __restrict__ warmp,
                        const float* __restrict__ gtbox, const float* __restrict__ gtt,
                        const int* __restrict__ gmeta, const float* __restrict__ glast,
                        float* __restrict__ p3) {
  __shared__ float swv[8][7];
  int tid = threadIdx.x;
  int e0 = blockIdx.x * 256 + tid;
  float valid = (e0 < NB*G_) ? 1.0f : 0.0f;
  int e = e0 < NB*G_ ? e0 : NB*G_ - 1;                  // clamp: keep EXEC full
  int b = e / G_;
  int warm = warmp[0];
  int gi = gmeta[e*4+0], gj = gmeta[e*4+1], a = gmeta[e*4+2], lb = gmeta[e*4+3];
  float txc = gtt[e*4+0], tyc = gtt[e*4+1], tw = gtt[e*4+2], th = gtt[e*4+3];
  float last = glast[e] * valid;
  const float* base = outp + (size_t)b*BSTRIDE + (size_t)a*CH*PLANE + gj*NW + gi;
  float x = base[0], y = base[PLANE], w = base[2*PLANE], h = base[3*PLANE], conf = base[4*PLANE];
  float sx = sigmoidf_(x), sy = sigmoidf_(y);
  float pw = __expf(w) * anchW(a), ph = __expf(h) * anchH(a);
  float px = sx + (float)gi, py = sy + (float)gj;
  float p0 = px - pw*0.5f, p1 = py - ph*0.5f, p2c = px + pw*0.5f, p3c = py + ph*0.5f;
  float ap = pw * ph;
  // nCorrect: 1-on-1 IoU vs own GT (counted for every (b,g), no dedup)
  float g0 = gtbox[e*5+0], g1 = gtbox[e*5+1], g2 = gtbox[e*5+2], g3 = gtbox[e*5+3], ag = gtbox[e*5+4];
  float iw = fmaxf(fminf(p2c, g2) - fmaxf(p0, g0), 0.0f);
  float ih = fmaxf(fminf(p3c, g3) - fmaxf(p1, g1), 0.0f);
  float inter = iw * ih;
  float iou1 = inter / (ap + ag - inter + 1e-10f);
  float c_corr = valid * (iou1 > 0.5f ? 1.0f : 0.0f);
  // base conf weight at this cell (recompute ignore mask)
  float miou = 0.0f;
  for (int g = 0; g < G_; ++g) {
    int eg = b * G_ + g;
    float q0 = gtbox[eg*5+0], q1 = gtbox[eg*5+1], q2 = gtbox[eg*5+2], q3 = gtbox[eg*5+3], qa = gtbox[eg*5+4];
    float iw2 = fmaxf(fminf(p2c, q2) - fmaxf(p0, q0), 0.0f);
    float ih2 = fmaxf(fminf(p3c, q3) - fmaxf(p1, q1), 0.0f);
    float in2 = iw2 * ih2;
    miou = fmaxf(miou, in2 / (ap + qa - in2 + 1e-10f));
  }
  float cwb = (miou > 0.5f) ? 0.0f : 0.5f;
  float spc = softplusf_(conf);
  float c_conf = last * (5.0f * (spc - conf) - cwb * spc);  // OBJ override minus base
  float dxc = (sx - txc) * (sx - txc);
  float dyc = (sy - tyc) * (sy - tyc);
  float dw  = (w - tw) * (w - tw);
  float dh  = (h - th) * (h - th);
  if (warm) {                                            // subtract warm-up base
    dxc -= (sx - 0.5f) * (sx - 0.5f);
    dyc -= (sy - 0.5f) * (sy - 0.5f);
    dw  -= w * w;
    dh  -= h * h;
  }
  float c_xc = last * dxc, c_yc = last * dyc, c_w = last * dw, c_h = last * dh;
  float c_cls = 0.0f;
  for (int c = 0; c < NC_; ++c) {
    float lg = base[(5 + c) * PLANE];
    c_cls += softplusf_(lg) - ((c == lb) ? lg : 0.0f);   // bce(lg, one_hot)
  }
  c_cls *= last;
  float s[7] = { wave_sum(c_xc), wave_sum(c_yc), wave_sum(c_w), wave_sum(c_h),
                 wave_sum(c_conf), wave_sum(c_cls), wave_sum(c_corr) };
  int wv = tid >> 5;
  if ((tid & 31) == 0) for (int k = 0; k < 7; ++k) swv[wv][k] = s[k];
  __syncthreads();
  if (tid == 0) {
    for (int k = 0; k < 7; ++k) {
      float t = 0.f;
      for (int wvi = 0; wvi < 8; ++wvi) t += swv[wvi][k];
      p3[k*NBLK3 + blockIdx.x] = t;
    }
  }
}

// ---------------- Kernel 4: final deterministic reduction ----------------
__global__ void k_final(const float* __restrict__ p2, const float* __restrict__ p3,
                        float* __restrict__ res) {
  int lane = threadIdx.x;  // one wave of 32
  float tot[7];
  for (int cmp = 0; cmp < 5; ++cmp) {
    float acc = 0.f;
    for (int k = lane; k < TOT2; k += 32) acc += p2[cmp*TOT2 + k];
    tot[cmp] = wave_sum(acc);
  }
  tot[5] = 0.f; tot[6] = 0.f;
  for (int cmp = 0; cmp < 7; ++cmp) {
    int idx = lane < NBLK3 ? lane : 0;                   // clamp: EXEC stays full
    float v = p3[cmp*NBLK3 + idx] * (lane < NBLK3 ? 1.0f : 0.0f);
    tot[cmp] += wave_sum(v);
  }
  if (lane == 0) {
    const float invB = 1.0f / (float)NB;
    float lxc = tot[0] * 0.5f * invB;
    float lyc = tot[1] * 0.5f * invB;
    float lw  = tot[2] * 0.5f * invB;
    float lh  = tot[3] * 0.5f * invB;
    float lcf = tot[4] * invB;
    float lcl = tot[5] * invB;
    res[0] = lxc + lyc + lw + lh + lcf + lcl;  // loss
    res[1] = (float)(NB * G_);                 // nGT
    res[2] = tot[6];                           // nCorrect
    res[3] = lxc; res[4] = lyc; res[5] = lw; res[6] = lh; res[7] = lcf; res[8] = lcl;
  }
}

extern "C" void kernel_launch(void* const* d_in, const int* in_sizes, int n_in,
                              void* d_out, int out_size, void* d_ws, size_t ws_size,
                              hipStream_t stream) {
  const float* out_t = (const float*)d_in[0];
  const float* bb    = (const float*)d_in[1];
  const int*   lab   = (const int*)d_in[2];
  const int*   aidx  = (const int*)d_in[3];
  const int*   warm  = (const int*)d_in[4];

  float* ws    = (float*)d_ws;
  float* gtbox = ws;                     // 640*5
  float* gtt   = gtbox + 640*5;          // 640*4
  int*   gmeta = (int*)(gtt + 640*4);    // 640*4 ints
  float* glast = (float*)(gmeta + 640*4);// 640
  float* p2    = glast + 640;            // 5*1024
  float* p3    = p2 + 5*TOT2;            // 7*3

  k_prep <<<(NB*G_ + 255)/256, 256, 0, stream>>>(bb, lab, aidx, gtbox, gtt, gmeta, glast);
  k_main <<<dim3(NBLK2, NB), 256, 0, stream>>>(out_t, warm, gtbox, p2);
  k_match<<<NBLK3, 256, 0, stream>>>(out_t, warm, gtbox, gtt, gmeta, glast, p3);
  k_final<<<1, 32, 0, stream>>>(p2, p3, (float*)d_out);
}